// P2RLossV9_59914793779730
// MI455X (gfx1250) — compile-verified
//
#include <hip/hip_runtime.h>
#include <stdint.h>

// Problem constants (match reference setup_inputs)
#define BB    64
#define HH    512
#define WW    512
#define NPTS  512
#define HWSZ  (HH*WW)
#define NENT  (NPTS*9)          // splat entries per batch (512 pts x 3x3)

// Workspace layout:
//   [0 .. 2KB)        : accum[BB*8] floats  {S1, S2, tsum, cross, tsq, pad..}
//   [4KB .. 4KB+64MB) : dense target map, BB*HH*WW floats (only touched cells used)
// Requires ws_size >= 4096 + (size_t)BB*HWSZ*4  (~64 MB + 4 KB)
#define ACC_STRIDE 8
#define MAP_OFFSET 4096

// ---------------------------------------------------------------------------
// Splat weight: exp(-sqrt(dy^2+dx^2)/2) takes only 3 values on the 3x3 stencil.
// Precomputed to f32 precision (matches jnp.exp float32 within 1 ulp) — avoids
// transcendental VALU ops in the sparse kernels.
// ---------------------------------------------------------------------------
__device__ __forceinline__ float splat_weight(int r2) {
    return (r2 == 0) ? 1.0f : ((r2 == 1) ? 0.60653066f : 0.49306870f);
}

// Map a (batch, entry) to its splat pixel + weight. Mirrors the reference:
// x = int(clip(px*scale, 0, W-1)); trunc == floor since operand >= 0, and
// scale = 512/2048 = 0.25 exactly (power of two), so no boundary rounding.
__device__ __forceinline__ bool entry_pixel(const float* __restrict__ pts,
                                            int b, int e, int Hin, int Win,
                                            int& idx, float& w) {
    int pt = e / 9;
    int o  = e - pt * 9;
    int dy = o / 3 - 1;
    int dx = o - (o / 3) * 3 - 1;
    const float* p = pts + (((size_t)b * NPTS + pt) << 1);
    float sw = (float)WW / (float)Win;
    float sh = (float)HH / (float)Hin;
    int x = (int)fminf(fmaxf(p[0] * sw, 0.0f), (float)(WW - 1));
    int y = (int)fminf(fmaxf(p[1] * sh, 0.0f), (float)(HH - 1));
    int nx = x + dx, ny = y + dy;
    if (nx < 0 || nx >= WW || ny < 0 || ny >= HH) { w = 0.0f; return false; }
    w = splat_weight(dy * dy + dx * dx);
    idx = ny * WW + nx;
    return true;
}

// 256-thread (8-wave, wave32) block reduction; result valid in thread 0.
__device__ __forceinline__ float blockReduceSum256(float v, float* lds) {
    #pragma unroll
    for (int m = 16; m; m >>= 1) v += __shfl_xor(v, m, 32);
    int lane = threadIdx.x & 31, wid = threadIdx.x >> 5;
    __syncthreads();                 // protect lds reuse across calls
    if (lane == 0) lds[wid] = v;
    __syncthreads();
    if (threadIdx.x == 0) {
        float s = 0.0f;
        #pragma unroll
        for (int i = 0; i < 8; ++i) s += lds[i];
        return s;
    }
    return 0.0f;
}

// ---------------------------------------------------------------------------
// K0: zero accumulators + zero only the map cells this call will splat.
// (d_ws is not re-poisoned between replays; zeroing only touched cells avoids
//  streaming 64 MB of zeros every call.)
// ---------------------------------------------------------------------------
__global__ __launch_bounds__(256) void k_init(const float* __restrict__ pts,
                                              const int* __restrict__ pHin,
                                              const int* __restrict__ pWin,
                                              float* __restrict__ accum,
                                              float* __restrict__ map) {
    int b = blockIdx.y;
    int e = blockIdx.x * 256 + threadIdx.x;
    if (b == 0 && blockIdx.x == 0) {
        accum[threadIdx.x] = 0.0f;
        accum[256 + threadIdx.x] = 0.0f;   // 512 accumulator slots total
    }
    if (e < NENT) {
        int idx; float w;
        if (entry_pixel(pts, b, e, *pHin, *pWin, idx, w))
            map[(size_t)b * HWSZ + idx] = 0.0f;   // racy identical stores: benign
    }
}

// ---------------------------------------------------------------------------
// K1: single pass over pred computing S1 = sum(p), S2 = sum(p^2) per batch.
// Memory-bound (64 MB total -> ~3 us floor at 23.3 TB/s). CDNA5 async
// global->LDS pipeline: each block stages 32 KB in two 16 KB stages; all 8
// per-thread B128 DMAs are issued up front (each fully coalesced: consecutive
// lanes -> consecutive float4s, 512 B/wave/instruction), then stage 0 is
// consumed behind a partial wait (async loads complete in order, so
// asynccnt<=4 => stage 0 resident) while stage 1 is still in flight.
// grid = (32, BB) = 2048 blocks x 8 waves; 32 KB LDS -> 10 blocks/WGP.
// ---------------------------------------------------------------------------
__global__ __launch_bounds__(256) void k_pred_reduce(const float* __restrict__ pred,
                                                     float* __restrict__ accum) {
    __shared__ __align__(16) float4 buf4[2048];     // 32 KB staging (2 stages)
    __shared__ float part[16];
    const int tid = threadIdx.x;
    const int b   = blockIdx.y;

    const float4* g4 = (const float4*)(pred + (size_t)b * HWSZ
                                            + (size_t)blockIdx.x * 8192);
    // LDS byte address (low 32 bits of flat addr == ds offset)
    const uint32_t ldsBase = (uint32_t)(uintptr_t)(&buf4[0]);

    // Issue both stages' async DMAs back-to-back (ASYNCcnt = 8 outstanding).
    #pragma unroll
    for (int s = 0; s < 2; ++s) {
        #pragma unroll
        for (int i = 0; i < 4; ++i) {
            int j = s * 1024 + i * 256 + tid;
            asm volatile("global_load_async_to_lds_b128 %0, %1, off"
                         :: "v"(ldsBase + (uint32_t)j * 16u), "v"(g4 + j)
                         : "memory");
        }
    }

    float s1 = 0.0f, s2 = 0.0f;

    // Stage 0: first 4 per-wave DMAs retired once ASYNCcnt <= 4 (in-order).
    asm volatile("s_wait_asynccnt 0x4" ::: "memory");
    #pragma unroll
    for (int i = 0; i < 4; ++i) {
        float4 v = buf4[i * 256 + tid];             // ds_load_b128
        s1 += v.x + v.y + v.z + v.w;
        s2 += v.x * v.x + v.y * v.y + v.z * v.z + v.w * v.w;
    }

    // Stage 1: drain remaining DMAs.
    asm volatile("s_wait_asynccnt 0x0" ::: "memory");
    #pragma unroll
    for (int i = 0; i < 4; ++i) {
        float4 v = buf4[1024 + i * 256 + tid];
        s1 += v.x + v.y + v.z + v.w;
        s2 += v.x * v.x + v.y * v.y + v.z * v.z + v.w * v.w;
    }

    #pragma unroll
    for (int m = 16; m; m >>= 1) {
        s1 += __shfl_xor(s1, m, 32);
        s2 += __shfl_xor(s2, m, 32);
    }
    int lane = tid & 31, wid = tid >> 5;
    __syncthreads();
    if (lane == 0) { part[wid] = s1; part[8 + wid] = s2; }
    __syncthreads();
    if (tid == 0) {
        float a = 0.0f, c = 0.0f;
        #pragma unroll
        for (int i = 0; i < 8; ++i) { a += part[i]; c += part[8 + i]; }
        atomicAdd(&accum[b * ACC_STRIDE + 0], a);
        atomicAdd(&accum[b * ACC_STRIDE + 1], c);
    }
}

// ---------------------------------------------------------------------------
// K2: splat entries into the sparse map, accumulate tsum[b] = sum of weights.
// ---------------------------------------------------------------------------
__global__ __launch_bounds__(256) void k_splat(const float* __restrict__ pts,
                                               const int* __restrict__ pHin,
                                               const int* __restrict__ pWin,
                                               float* __restrict__ accum,
                                               float* __restrict__ map) {
    __shared__ float lds[8];
    int b = blockIdx.y;
    int e = blockIdx.x * 256 + threadIdx.x;
    float w = 0.0f; int idx;
    if (e < NENT && entry_pixel(pts, b, e, *pHin, *pWin, idx, w))
        atomicAdd(&map[(size_t)b * HWSZ + idx], w);
    float t = blockReduceSum256(w, lds);
    if (threadIdx.x == 0) atomicAdd(&accum[b * ACC_STRIDE + 2], t);
}

// ---------------------------------------------------------------------------
// K3: sparse gather. cross = sum_e w*pred[pix];  tsq = sum_e w*T[pix]
// (sum_e w*T equals sum_pixels T^2 exactly — handles duplicate splats).
// pred lines were just streamed by K1 with RT hint -> hits in 192 MB L2.
// ---------------------------------------------------------------------------
__global__ __launch_bounds__(256) void k_gather(const float* __restrict__ pred,
                                                const float* __restrict__ pts,
                                                const int* __restrict__ pHin,
                                                const int* __restrict__ pWin,
                                                const float* __restrict__ map,
                                                float* __restrict__ accum) {
    __shared__ float lds[8];
    int b = blockIdx.y;
    int e = blockIdx.x * 256 + threadIdx.x;
    float cr = 0.0f, tq = 0.0f;
    int idx; float w;
    if (e < NENT && entry_pixel(pts, b, e, *pHin, *pWin, idx, w)) {
        float T = map[(size_t)b * HWSZ + idx];
        float p = pred[(size_t)b * HWSZ + idx];
        cr = w * p;
        tq = w * T;
    }
    float c = blockReduceSum256(cr, lds);
    float t = blockReduceSum256(tq, lds);
    if (threadIdx.x == 0) {
        atomicAdd(&accum[b * ACC_STRIDE + 3], c);
        atomicAdd(&accum[b * ACC_STRIDE + 4], t);
    }
}

// ---------------------------------------------------------------------------
// K4: combine per-batch partials into (total, count_loss, spatial_loss).
// spatial/batch = (S2/pe^2 - 2*cross/(pe*ts) + tsq/ts^2) / (H*W),  pe = S1+1e-8
// ---------------------------------------------------------------------------
__global__ __launch_bounds__(64) void k_final(const float* __restrict__ accum,
                                              const int* __restrict__ pcell,
                                              float* __restrict__ out) {
    __shared__ float sc[64], ss[64];
    int b = threadIdx.x;
    float cl = 0.0f, sl = 0.0f;
    if (b < BB) {
        float S1 = accum[b * ACC_STRIDE + 0];
        float S2 = accum[b * ACC_STRIDE + 1];
        float ts = accum[b * ACC_STRIDE + 2];
        float cr = accum[b * ACC_STRIDE + 3];
        float tq = accum[b * ACC_STRIDE + 4];
        float cell = (float)(*pcell);
        cl = fabsf(S1 / cell - (float)NPTS);
        float pe = S1 + 1e-8f;
        float sp = S2 / (pe * pe) - 2.0f * cr / (pe * ts) + tq / (ts * ts);
        sl = sp / (float)HWSZ;
    }
    sc[threadIdx.x] = cl; ss[threadIdx.x] = sl;
    __syncthreads();
    if (threadIdx.x == 0) {
        float c = 0.0f, s = 0.0f;
        for (int i = 0; i < BB; ++i) { c += sc[i]; s += ss[i]; }
        c /= (float)BB; s /= (float)BB;
        out[0] = 2.5f * c + 0.1f * s;   // COUNT_W, SPATIAL_W (SCALE term is 0)
        out[1] = c;
        out[2] = s;
    }
}

// ---------------------------------------------------------------------------
// Inputs (setup_inputs order): pred f32, points f32, cell_area i32[1],
// H_in i32[1], W_in i32[1].  Output: 3 f32 scalars (total, count, spatial).
// ---------------------------------------------------------------------------
extern "C" void kernel_launch(void* const* d_in, const int* in_sizes, int n_in,
                              void* d_out, int out_size, void* d_ws, size_t ws_size,
                              hipStream_t stream) {
    const float* pred = (const float*)d_in[0];
    const float* pts  = (const float*)d_in[1];
    const int*   cell = (const int*)d_in[2];
    const int*   Hin  = (const int*)d_in[3];
    const int*   Win  = (const int*)d_in[4];
    float* out   = (float*)d_out;
    float* accum = (float*)d_ws;
    float* map   = (float*)((char*)d_ws + MAP_OFFSET);

    dim3 gE((NENT + 255) / 256, BB);     // 18 x 64 blocks over splat entries
    dim3 gR(32, BB);                     // 32 x 64 blocks over pred (8192 f32/block)

    k_init       <<<gE, 256, 0, stream>>>(pts, Hin, Win, accum, map);
    k_pred_reduce<<<gR, 256, 0, stream>>>(pred, accum);
    k_splat      <<<gE, 256, 0, stream>>>(pts, Hin, Win, accum, map);
    k_gather     <<<gE, 256, 0, stream>>>(pred, pts, Hin, Win, map, accum);
    k_final      <<<1, 64, 0, stream>>>(accum, cell, out);
}